// HashEmbedder_8211977470231
// MI455X (gfx1250) — compile-verified
//
#include <hip/hip_runtime.h>

// ---------------------------------------------------------------------------
// HashEmbedder lookup (instant-NGP style hash grid, gather-only, no interp)
//
// out[n, l*2 + f] = tables[l, hash(floor(x[n]*res[l])) & (2^17-1), f]
//
// Memory-bound: ~450 MB HBM traffic (x stream + output stream); 12.6 MB of
// tables are L2-resident on MI455X (192 MB L2). Strategy:
//   * thread = (point, level) pair -> output stores are perfectly coalesced
//     (wave32 lanes write 32 consecutive float2's = one 256B burst)
//   * non-temporal stores for the 402 MB output so it does not evict the
//     L2-resident tables
//   * default (RT) temporal policy for table gathers -> they stay in L2
//   * global_prefetch_b8 (gfx1250, WGP scope) walks the x stream ahead
// ---------------------------------------------------------------------------

typedef __attribute__((ext_vector_type(2))) float v2f;

#define NUM_LEVELS   12
#define LOG2_HASH    17
#define HASH_MASK    ((1u << LOG2_HASH) - 1u)
#define PRIME_Y      2654435761u
#define PRIME_Z      805459861u

// RESOLUTIONS[i] = int(16 * (1024/16)**(i/11)), stored as f32 (matches reference)
__device__ __constant__ float kRes[NUM_LEVELS] = {
    16.0f, 23.0f, 34.0f, 49.0f, 72.0f, 105.0f,
    154.0f, 225.0f, 329.0f, 480.0f, 701.0f, 1024.0f
};

__global__ __launch_bounds__(256, 8)
void hash_embed_kernel(const float* __restrict__ x,       // [N, 3]
                       const v2f*   __restrict__ tables,  // [12, 131072] float2
                       v2f*         __restrict__ out,     // [N, 12] float2
                       unsigned total,                    // N * 12
                       unsigned n_points)
{
    const unsigned tid = blockIdx.x * blockDim.x + threadIdx.x;
    if (tid >= total) return;

    const unsigned n = tid / NUM_LEVELS;        // point index
    const unsigned l = tid - n * NUM_LEVELS;    // level index

    // gfx1250: speculative WGP-scope prefetch of the x stream ~96 KB ahead
    // (one lane per wave; speculative prefetches are dropped if OOB/unmapped)
    if ((threadIdx.x & 31u) == 0u) {
        const unsigned npf = n + 8192u;
        if (npf < n_points) {
            __builtin_prefetch(&x[3u * npf], 0, 3);   // -> global_prefetch_b8 (WGP)
        }
    }

    // load point (3 contiguous floats -> global_load_b96)
    const float* px = x + 3u * n;
    const float fx = px[0];
    const float fy = px[1];
    const float fz = px[2];

    // scale by this level's resolution; floor == trunc for nonnegative domain
    const float r = kRes[l];
    const unsigned x0 = (unsigned)(fx * r);
    const unsigned y0 = (unsigned)(fy * r);
    const unsigned z0 = (unsigned)(fz * r);

    // spatial hash: uint32 wraparound multiply, XOR, low-17-bit mask
    const unsigned h = (x0 ^ (y0 * PRIME_Y) ^ (z0 * PRIME_Z)) & HASH_MASK;

    // gather 2 features (global_load_b64); tables stay RT -> L2-resident
    const v2f f = tables[(l << LOG2_HASH) + h];

    // out float2 index == n*12 + l == tid; lanes write consecutive addresses.
    // Non-temporal store: stream the 402 MB output past L2 (keep L2 for tables)
    __builtin_nontemporal_store(f, &out[tid]);
}

extern "C" void kernel_launch(void* const* d_in, const int* in_sizes, int n_in,
                              void* d_out, int out_size, void* d_ws, size_t ws_size,
                              hipStream_t stream)
{
    const float* x      = (const float*)d_in[0];   // [N,3] f32
    const v2f*   tables = (const v2f*)  d_in[1];   // [12, 131072, 2] f32
    v2f*         out    = (v2f*)d_out;             // [N, 24] f32 == [N,12] float2

    const unsigned n_points = (unsigned)(in_sizes[0] / 3);
    const unsigned total    = n_points * NUM_LEVELS;

    const int threads = 256;
    const unsigned blocks = (total + threads - 1) / threads;

    hash_embed_kernel<<<dim3(blocks), dim3(threads), 0, stream>>>(
        x, tables, out, total, n_points);
}